// BoundaryEncoder_82832739271130
// MI455X (gfx1250) — compile-verified
//
#include <hip/hip_runtime.h>
#include <math.h>

#define D_FEAT 128
#define ED_FEAT 32
#define NEG_SLOPE 0.2f

typedef __attribute__((ext_vector_type(2))) float v2f;
typedef __attribute__((ext_vector_type(8))) float v8f;

// ---------------------------------------------------------------------------
// GEMM: XT = X @ W   (M = n nodes, K = N = 128) using V_WMMA_F32_16X16X4_F32.
// Block = 256 threads = 8 waves; block computes a 16x128 row-tile; wave w
// computes the 16x16 tile at columns [16w,16w+16). A-tile staged in LDS.
// ---------------------------------------------------------------------------
__global__ __launch_bounds__(256)
void gemm_xt_kernel(const float* __restrict__ X, const float* __restrict__ W,
                    float* __restrict__ XT, int n) {
  __shared__ float sA[16 * D_FEAT];  // 8 KB
  const int tid = threadIdx.x;
  const int r0 = blockIdx.x * 16;
  for (int i = tid; i < 16 * D_FEAT; i += 256) {
    int r = i >> 7, c = i & 127;
    sA[i] = (r0 + r < n) ? X[(size_t)(r0 + r) * D_FEAT + c] : 0.0f;
  }
  __syncthreads();
  const int wave = tid >> 5;
  const int lane = tid & 31;
  const int c0 = wave * 16;
  const int m = lane & 15;          // M (for A) / N (for B) within tile
  const int kh = (lane >> 4) << 1;  // lanes 0-15 -> K 0,1 ; lanes 16-31 -> K 2,3
  v8f acc = {};
  for (int kb = 0; kb < D_FEAT; kb += 4) {
    const int k0 = kb + kh;
    v2f a, b;
    a.x = sA[m * D_FEAT + k0];
    a.y = sA[m * D_FEAT + k0 + 1];
    b.x = W[(size_t)k0 * D_FEAT + c0 + m];
    b.y = W[(size_t)(k0 + 1) * D_FEAT + c0 + m];
    acc = __builtin_amdgcn_wmma_f32_16x16x4_f32(false, a, false, b, (short)0,
                                                acc, false, false);
  }
  // C/D layout: VGPR g, lanes 0-15 -> M=g ; lanes 16-31 -> M=g+8 ; N = lane&15
  const int row_off = (lane >> 4) * 8;
  const int col = c0 + m;
  for (int g = 0; g < 8; ++g) {
    int row = r0 + g + row_off;
    if (row < n) XT[(size_t)row * D_FEAT + col] = acc[g];
  }
}

// ---------------------------------------------------------------------------
// Per-node attention terms: a_src[i] = xt[i]·as ; a_dst[i] = xt[i]·ad
// One wave (32 lanes) per node.
// ---------------------------------------------------------------------------
__global__ __launch_bounds__(256)
void node_alpha_kernel(const float* __restrict__ XT, const float* __restrict__ as_,
                       const float* __restrict__ ad_, float* __restrict__ osrc,
                       float* __restrict__ odst, int n) {
  const int w = (blockIdx.x * blockDim.x + threadIdx.x) >> 5;
  const int lane = threadIdx.x & 31;
  if (w >= n) return;
  const float* row = XT + (size_t)w * D_FEAT;
  float ss = 0.f, sd = 0.f;
  for (int j = lane; j < D_FEAT; j += 32) {
    float v = row[j];
    ss += v * as_[j];
    sd += v * ad_[j];
  }
  for (int off = 16; off; off >>= 1) {
    ss += __shfl_down(ss, off, 32);
    sd += __shfl_down(sd, off, 32);
  }
  if (lane == 0) { osrc[w] = ss; odst[w] = sd; }
}

// w[k] = sum_j We[k][j] * ae[j]   (32 outputs)
__global__ void compute_w_kernel(const float* __restrict__ We,
                                 const float* __restrict__ ae,
                                 float* __restrict__ wv) {
  int k = threadIdx.x;
  if (k < ED_FEAT) {
    float s = 0.f;
    for (int j = 0; j < D_FEAT; ++j) s += We[(size_t)k * D_FEAT + j] * ae[j];
    wv[k] = s;
  }
}

// Per real edge: eterm[e] = edge_attr[e]·w ; accumulate per-dst sum and count
// (used to synthesize the self-loop edge-attr term, which is linear in attr).
__global__ __launch_bounds__(256)
void eterm_kernel(const float* __restrict__ EA, const int* __restrict__ dst,
                  const float* __restrict__ wv, float* __restrict__ eterm,
                  float* __restrict__ loop_acc, float* __restrict__ cnt, int E) {
  int e = blockIdx.x * blockDim.x + threadIdx.x;
  if (e >= E) return;
  const float* a = EA + (size_t)e * ED_FEAT;
  float s = 0.f;
  for (int j = 0; j < ED_FEAT; ++j) s += a[j] * wv[j];
  eterm[e] = s;
  int d = dst[e];
  atomicAdd(&loop_acc[d], s);
  atomicAdd(&cnt[d], 1.0f);
}

__global__ void loop_div_kernel(float* __restrict__ loop_acc,
                                const float* __restrict__ cnt, int n) {
  int i = blockIdx.x * blockDim.x + threadIdx.x;
  if (i < n) loop_acc[i] = loop_acc[i] / fmaxf(cnt[i], 1.0f);
}

// Monotonic float <-> uint encoding for atomicMax on signed floats.
__device__ __forceinline__ unsigned enc_f(float f) {
  unsigned u = __float_as_uint(f);
  return (u & 0x80000000u) ? ~u : (u | 0x80000000u);
}
__device__ __forceinline__ float dec_f(unsigned u) {
  u = (u & 0x80000000u) ? (u & 0x7FFFFFFFu) : ~u;
  return __uint_as_float(u);
}

// Pass 1 over [0,E) real edges and [E,E+n) self loops:
// logit = leaky_relu(a_src[s] + a_dst[d] + eterm) ; segment max via atomicMax.
__global__ __launch_bounds__(256)
void pass1_kernel(const int* __restrict__ src, const int* __restrict__ dst,
                  const float* __restrict__ eterm, const float* __restrict__ loop_et,
                  const float* __restrict__ asv, const float* __restrict__ adv,
                  float* __restrict__ logit, unsigned* __restrict__ menc,
                  int E, int n) {
  int i = blockIdx.x * blockDim.x + threadIdx.x;
  if (i >= E + n) return;
  int s, d; float et;
  if (i < E) { s = src[i]; d = dst[i]; et = eterm[i]; }
  else       { s = d = i - E;          et = loop_et[s]; }
  float x = asv[s] + adv[d] + et;
  x = (x > 0.f) ? x : NEG_SLOPE * x;
  logit[i] = x;
  atomicMax(&menc[d], enc_f(x));
}

// Pass 2: p = exp(logit - max[dst]) ; segment sum via atomicAdd.
__global__ __launch_bounds__(256)
void pass2_kernel(const int* __restrict__ dst, const float* __restrict__ logit,
                  const unsigned* __restrict__ menc, float* __restrict__ p,
                  float* __restrict__ denom, int E, int n) {
  int i = blockIdx.x * blockDim.x + threadIdx.x;
  if (i >= E + n) return;
  int d = (i < E) ? dst[i] : (i - E);
  float pv = expf(logit[i] - dec_f(menc[d]));
  p[i] = pv;
  atomicAdd(&denom[d], pv);
}

// Pass 3: out[dst] += (p/denom[dst]) * xt[src]   — one wave per (edge|loop),
// 32 lanes x float4 = 128 columns.
__global__ __launch_bounds__(256)
void pass3_kernel(const int* __restrict__ src, const int* __restrict__ dst,
                  const float* __restrict__ p, const float* __restrict__ denom,
                  const float* __restrict__ XT, float* __restrict__ out,
                  int E, int n) {
  long long wi = ((long long)blockIdx.x * blockDim.x + threadIdx.x) >> 5;
  const int lane = threadIdx.x & 31;
  if (wi >= (long long)E + n) return;
  int s, d;
  if (wi < E) { s = src[wi]; d = dst[wi]; } else { s = d = (int)(wi - E); }
  float coeff = p[wi] / fmaxf(denom[d], 1e-16f);
  const float4 v = ((const float4*)(XT + (size_t)s * D_FEAT))[lane];
  float* orow = out + (size_t)d * D_FEAT + 4 * lane;
  atomicAdd(&orow[0], coeff * v.x);
  atomicAdd(&orow[1], coeff * v.y);
  atomicAdd(&orow[2], coeff * v.z);
  atomicAdd(&orow[3], coeff * v.w);
}

__global__ __launch_bounds__(256)
void epilogue_kernel(float* __restrict__ x, const float* __restrict__ b,
                     int n, int do_relu) {
  long long i = (long long)blockIdx.x * blockDim.x + threadIdx.x;
  if (i >= (long long)n * D_FEAT) return;
  float v = x[i] + b[i & (D_FEAT - 1)];
  if (do_relu) v = fmaxf(v, 0.f);
  x[i] = v;
}

// Column sums over nodes: <<<128,128>>>; thread t of block b sums column t over
// rows b, b+128, ... then one atomicAdd per (block, column).
__global__ __launch_bounds__(128)
void pool_kernel(const float* __restrict__ x, float* __restrict__ colsum, int n) {
  const int t = threadIdx.x;
  float s = 0.f;
  for (int r = blockIdx.x; r < n; r += gridDim.x)
    s += x[(size_t)r * D_FEAT + t];
  atomicAdd(&colsum[t], s);
}

__global__ __launch_bounds__(128)
void ln_kernel(const float* __restrict__ colsum, const float* __restrict__ gamma,
               const float* __restrict__ beta, float* __restrict__ out, float inv_n) {
  __shared__ float red[D_FEAT];
  const int t = threadIdx.x;
  float v = colsum[t] * inv_n;  // pooled[t]
  red[t] = v; __syncthreads();
  for (int s = 64; s; s >>= 1) { if (t < s) red[t] += red[t + s]; __syncthreads(); }
  float mu = red[0] / (float)D_FEAT;
  __syncthreads();
  float dv = (v - mu) * (v - mu);
  red[t] = dv; __syncthreads();
  for (int s = 64; s; s >>= 1) { if (t < s) red[t] += red[t + s]; __syncthreads(); }
  float var = red[0] / (float)D_FEAT;
  out[t] = (v - mu) * rsqrtf(var + 1e-5f) * gamma[t] + beta[t];
}

// ---------------------------------------------------------------------------
extern "C" void kernel_launch(void* const* d_in, const int* in_sizes, int n_in,
                              void* d_out, int out_size, void* d_ws, size_t ws_size,
                              hipStream_t stream) {
  const float* bnd_nodes = (const float*)d_in[0];
  const int*   edge_idx  = (const int*)d_in[1];
  const float* edge_attr = (const float*)d_in[2];
  const float* Ws        = (const float*)d_in[3];
  const float* att_src   = (const float*)d_in[4];
  const float* att_dst   = (const float*)d_in[5];
  const float* We        = (const float*)d_in[6];
  const float* att_edge  = (const float*)d_in[7];
  const float* biases    = (const float*)d_in[8];
  const float* ln_gamma  = (const float*)d_in[9];
  const float* ln_beta   = (const float*)d_in[10];
  float* out = (float*)d_out;

  const int N = in_sizes[0] / D_FEAT;
  const int E = in_sizes[1] / 2;
  const int L = in_sizes[3] / (D_FEAT * D_FEAT);
  const int* src = edge_idx;
  const int* dst = edge_idx + E;

  // -------- workspace layout --------
  char* base = (char*)d_ws;
  size_t off = 0;
  auto alloc = [&](size_t bytes) -> char* {
    char* p = base + off;
    off = (off + bytes + 255) & ~(size_t)255;
    return p;
  };
  float*    xt       = (float*)alloc((size_t)N * D_FEAT * 4);
  float*    xbuf     = (float*)alloc((size_t)N * D_FEAT * 4);  // layer io + accum
  float*    a_src_v  = (float*)alloc((size_t)N * 4);
  float*    a_dst_v  = (float*)alloc((size_t)N * 4);
  float*    wvec     = (float*)alloc(ED_FEAT * 4);
  float*    eterm    = (float*)alloc((size_t)E * 4);
  float*    loop_acc = (float*)alloc((size_t)N * 4);
  float*    cnt      = (float*)alloc((size_t)N * 4);
  unsigned* menc     = (unsigned*)alloc((size_t)N * 4);
  float*    denom    = (float*)alloc((size_t)N * 4);
  float*    logit    = (float*)alloc((size_t)(E + N) * 4);
  float*    pbuf     = (float*)alloc((size_t)(E + N) * 4);
  float*    colsum   = (float*)alloc(D_FEAT * 4);
  (void)ws_size; (void)n_in; (void)out_size;

  const int row_tiles = (N + 15) / 16;
  const int EN = E + N;

  for (int l = 0; l < L; ++l) {
    const float* x_in = (l == 0) ? bnd_nodes : xbuf;

    gemm_xt_kernel<<<row_tiles, 256, 0, stream>>>(
        x_in, Ws + (size_t)l * D_FEAT * D_FEAT, xt, N);

    // xbuf now becomes the output accumulator (gemm already consumed it)
    hipMemsetAsync(xbuf, 0, (size_t)N * D_FEAT * 4, stream);
    hipMemsetAsync(loop_acc, 0, (size_t)N * 4, stream);
    hipMemsetAsync(cnt, 0, (size_t)N * 4, stream);
    hipMemsetAsync(menc, 0, (size_t)N * 4, stream);
    hipMemsetAsync(denom, 0, (size_t)N * 4, stream);

    node_alpha_kernel<<<((size_t)N * 32 + 255) / 256, 256, 0, stream>>>(
        xt, att_src + (size_t)l * D_FEAT, att_dst + (size_t)l * D_FEAT,
        a_src_v, a_dst_v, N);

    compute_w_kernel<<<1, ED_FEAT, 0, stream>>>(
        We + (size_t)l * ED_FEAT * D_FEAT, att_edge + (size_t)l * D_FEAT, wvec);

    eterm_kernel<<<(E + 255) / 256, 256, 0, stream>>>(
        edge_attr, dst, wvec, eterm, loop_acc, cnt, E);

    loop_div_kernel<<<(N + 255) / 256, 256, 0, stream>>>(loop_acc, cnt, N);

    pass1_kernel<<<(EN + 255) / 256, 256, 0, stream>>>(
        src, dst, eterm, loop_acc, a_src_v, a_dst_v, logit, menc, E, N);

    pass2_kernel<<<(EN + 255) / 256, 256, 0, stream>>>(
        dst, logit, menc, pbuf, denom, E, N);

    pass3_kernel<<<((size_t)EN * 32 + 255) / 256, 256, 0, stream>>>(
        src, dst, pbuf, denom, xt, xbuf, E, N);

    epilogue_kernel<<<((size_t)N * D_FEAT + 255) / 256, 256, 0, stream>>>(
        xbuf, biases + (size_t)l * D_FEAT, N, (l < L - 1) ? 1 : 0);
  }

  hipMemsetAsync(colsum, 0, D_FEAT * 4, stream);
  pool_kernel<<<128, 128, 0, stream>>>(xbuf, colsum, N);
  ln_kernel<<<1, 128, 0, stream>>>(colsum, ln_gamma, ln_beta, out, 1.0f / (float)N);
}